// VectorReLU_63007170232699
// MI455X (gfx1250) — compile-verified
//
#include <hip/hip_runtime.h>
#include <hip/hip_bf16.h>
#include <stdint.h>

// Problem constants from the reference: x = (8, 4096, 4, 3, 128) f32
constexpr int B      = 8;
constexpr int M      = 16384;          // 4096 * 4
constexpr int D      = 128;
constexpr int K_RANK = M / 10;         // 1638 (1-indexed kth smallest)

// --------------------------------------------------------------------------
// Kernel 1: norms + transpose.  x:(B,M,3,D) -> normsT:(B*D, M) contiguous cols
// Global traffic fully vectorized to b128; LDS tile pitch 129 (conflict-free).
// --------------------------------------------------------------------------
__global__ __launch_bounds__(256) void norms_transpose_kernel(
    const float* __restrict__ x, float* __restrict__ normsT) {
  constexpr int MT = 64;
  __shared__ float tile[MT][D + 1];     // pitch 129 floats
  const int b   = blockIdx.x / (M / MT);
  const int m0  = (blockIdx.x % (M / MT)) * MT;
  const int tid = threadIdx.x;

  // Load phase: 3 x float4 per thread, coalesced along d
  for (int idx = tid; idx < MT * (D / 4); idx += 256) {
    const int ml = idx / (D / 4);
    const int d4 = idx % (D / 4);
    const size_t base =
        (((size_t)b * M + (m0 + ml)) * 3) * (size_t)D + 4 * d4;
    const float4 a = *(const float4*)(x + base);
    const float4 c = *(const float4*)(x + base + D);
    const float4 e = *(const float4*)(x + base + 2 * D);
    tile[ml][4 * d4 + 0] = sqrtf(a.x * a.x + c.x * c.x + e.x * e.x);
    tile[ml][4 * d4 + 1] = sqrtf(a.y * a.y + c.y * c.y + e.y * e.y);
    tile[ml][4 * d4 + 2] = sqrtf(a.z * a.z + c.z * c.z + e.z * e.z);
    tile[ml][4 * d4 + 3] = sqrtf(a.w * a.w + c.w * c.w + e.w * e.w);
  }
  __syncthreads();

  // Store phase: float4 along m (transposed), 16B-aligned, coalesced
  for (int idx = tid; idx < D * (MT / 4); idx += 256) {
    const int d  = idx / (MT / 4);
    const int ml = 4 * (idx % (MT / 4));
    float4 v;
    v.x = tile[ml + 0][d];
    v.y = tile[ml + 1][d];
    v.z = tile[ml + 2][d];
    v.w = tile[ml + 3][d];
    *(float4*)(normsT + ((size_t)b * D + d) * (size_t)M + (m0 + ml)) = v;
  }
}

// --------------------------------------------------------------------------
// Kernel 2: per-column exact k-th smallest via LDS radix select.
// Column (64 KB) is staged into LDS with CDNA5 async global->LDS DMA
// (GLOBAL_LOAD_ASYNC_TO_LDS_B128, tracked by ASYNCcnt).
// Norms are >= 0, so u32 bit-pattern order == float order.
// --------------------------------------------------------------------------
__global__ __launch_bounds__(256) void radix_select_kernel(
    const float* __restrict__ normsT, float* __restrict__ kx) {
  __shared__ unsigned int keys[M];      // 64 KB (WGP has 320 KB LDS)
  __shared__ unsigned int hist[256];
  __shared__ unsigned int s_pref;
  __shared__ int s_k;

  const int col = blockIdx.x;           // 0 .. B*D-1
  const int tid = threadIdx.x;
  const float* src = normsT + (size_t)col * M;

  // Flat address of a __shared__ object: addr[31:0] is the LDS byte offset.
  const unsigned int lds0 = (unsigned int)(uintptr_t)&keys[0];
  for (int i = tid; i < M / 4; i += 256) {
    const unsigned int lds_off = lds0 + (unsigned int)i * 16u;
    const float* g = src + (size_t)i * 4;
    asm volatile("global_load_async_to_lds_b128 %0, %1, off"
                 :
                 : "v"(lds_off), "v"(g)
                 : "memory");
  }
  asm volatile("s_wait_asynccnt 0x0" ::: "memory");
  __syncthreads();

  if (tid == 0) { s_pref = 0u; s_k = K_RANK; }
  __syncthreads();

  // 4 rounds x 8 bits radix select, entirely in LDS
  for (int shift = 24; shift >= 0; shift -= 8) {
    hist[tid] = 0u;                     // blockDim.x == 256
    __syncthreads();

    const unsigned int himask =
        (unsigned int)(0xFFFFFFFFull << (shift + 8));  // 0 when shift==24
    const unsigned int prefb = s_pref;

    for (int i = tid; i < M; i += 256) {
      const unsigned int key = keys[i];
      if ((key & himask) == prefb)
        atomicAdd(&hist[(key >> shift) & 0xFFu], 1u);
    }
    __syncthreads();

    if (tid == 0) {
      unsigned int cum = 0;
      const unsigned int kk = (unsigned int)s_k;
      int bsel = 0;
      for (; bsel < 256; ++bsel) {
        const unsigned int c = hist[bsel];
        if (cum + c >= kk) break;
        cum += c;
      }
      s_pref = prefb | ((unsigned int)bsel << shift);
      s_k    = (int)(kk - cum);
    }
    __syncthreads();
  }

  if (tid == 0) kx[col] = __uint_as_float(s_pref);
}

// --------------------------------------------------------------------------
// Kernel 3: threshold.  Recompute norm from x (x must be read anyway) and
// zero the 3-vector where norm <= kx[b,d].  4 d-channels per thread, all
// global traffic as b128.
// --------------------------------------------------------------------------
__global__ __launch_bounds__(256) void threshold_kernel(
    const float* __restrict__ x, const float* __restrict__ kx,
    float* __restrict__ out) {
  constexpr int D4 = D / 4;                                   // 32
  const size_t t = (size_t)blockIdx.x * 256 + threadIdx.x;    // over B*M*D4
  if (t >= (size_t)B * M * D4) return;
  const int    d4 = (int)(t % D4);
  const size_t md = t / D4;
  const int    m  = (int)(md % M);
  const int    b  = (int)(md / M);

  const size_t base = (((size_t)b * M + m) * 3) * (size_t)D + 4 * d4;
  const float4 a = *(const float4*)(x + base);
  const float4 c = *(const float4*)(x + base + D);
  const float4 e = *(const float4*)(x + base + 2 * D);
  const float4 thr = *(const float4*)(kx + b * D + 4 * d4);

  const bool k0 = sqrtf(a.x * a.x + c.x * c.x + e.x * e.x) > thr.x;
  const bool k1 = sqrtf(a.y * a.y + c.y * c.y + e.y * e.y) > thr.y;
  const bool k2 = sqrtf(a.z * a.z + c.z * c.z + e.z * e.z) > thr.z;
  const bool k3 = sqrtf(a.w * a.w + c.w * c.w + e.w * e.w) > thr.w;

  float4 o0, o1, o2;
  o0.x = k0 ? a.x : 0.0f;  o1.x = k0 ? c.x : 0.0f;  o2.x = k0 ? e.x : 0.0f;
  o0.y = k1 ? a.y : 0.0f;  o1.y = k1 ? c.y : 0.0f;  o2.y = k1 ? e.y : 0.0f;
  o0.z = k2 ? a.z : 0.0f;  o1.z = k2 ? c.z : 0.0f;  o2.z = k2 ? e.z : 0.0f;
  o0.w = k3 ? a.w : 0.0f;  o1.w = k3 ? c.w : 0.0f;  o2.w = k3 ? e.w : 0.0f;

  *(float4*)(out + base)         = o0;
  *(float4*)(out + base + D)     = o1;
  *(float4*)(out + base + 2 * D) = o2;
}

// --------------------------------------------------------------------------
extern "C" void kernel_launch(void* const* d_in, const int* in_sizes, int n_in,
                              void* d_out, int out_size, void* d_ws,
                              size_t ws_size, hipStream_t stream) {
  (void)in_sizes; (void)n_in; (void)out_size; (void)ws_size;
  const float* x = (const float*)d_in[0];
  float* out = (float*)d_out;

  // Workspace layout: normsT (B*D*M floats = 64 MiB) | kx (B*D floats)
  float* normsT = (float*)d_ws;
  float* kx     = (float*)((char*)d_ws + (size_t)B * D * M * sizeof(float));

  const int grid1 = B * (M / 64);                    // 2048 blocks
  const int grid2 = B * D;                           // 1024 blocks
  const int grid3 = (int)((size_t)B * M * (D / 4) / 256);  // 16384 blocks

  norms_transpose_kernel<<<grid1, 256, 0, stream>>>(x, normsT);
  radix_select_kernel<<<grid2, 256, 0, stream>>>(normsT, kx);
  threshold_kernel<<<grid3, 256, 0, stream>>>(x, kx, out);
}